// ReactantCentreIdentify_26723286516086
// MI455X (gfx1250) — compile-verified
//
#include <hip/hip_runtime.h>

// ext-vector types matching V_WMMA_F32_16X16X4_F32 operands:
// A: 16x4 f32 -> 2 VGPRs/lane, B: 4x16 f32 -> 2 VGPRs/lane, C/D: 16x16 f32 -> 8 VGPRs/lane.
typedef __attribute__((ext_vector_type(2))) float v2f;
typedef __attribute__((ext_vector_type(4))) float f4;
typedef __attribute__((ext_vector_type(8))) float v8f;

// ---------------------------------------------------------------------------
// Kernel 1: CSR bounds. batch[] is sorted, so rows of graph g are
// [lower_bound(batch,g), lower_bound(batch,g+1)). (B+1) binary searches.
// ---------------------------------------------------------------------------
__global__ void graph_bounds_kernel(const int* __restrict__ batch,
                                    const int* __restrict__ numg,
                                    int N, int* __restrict__ start) {
  const int B = *numg;
  for (int g = blockIdx.x * blockDim.x + threadIdx.x; g <= B;
       g += gridDim.x * blockDim.x) {
    int lo = 0, hi = N;
    while (lo < hi) {
      int mid = (lo + hi) >> 1;
      if (batch[mid] < g) lo = mid + 1; else hi = mid;
    }
    start[g] = lo;
  }
}

// ---------------------------------------------------------------------------
// Per-wave 16-dim chunk processor. FULL=true: all 16 dims valid -> no guards
// in the steady-state loop (unconditional NT loads/stores, running pointers).
// The <=3-row tail is peeled; its loads clamp the row (mask weight is 0, so
// the clamped value cannot contribute to the sum) so EXEC stays full at the
// WMMA. Labels are block-uniform -> scalar loads. Chunk-0 waves accumulate
// the condition-node count for free (countf = 1).
// ---------------------------------------------------------------------------
template <bool FULL>
__device__ __forceinline__ void process_chunk(
    const float* __restrict__ X, const int* __restrict__ label,
    float* __restrict__ out, float* __restrict__ pool,
    int s, int e, int D, int D2, int dim, int dimc, bool dok,
    int half, float countf, float& cntf) {
  v8f acc = {};
  const int     efull = s + ((e - s) & ~3);
  const size_t  stepX = (size_t)D * 4;
  const size_t  stepO = (size_t)D2 * 4;
  const float*  pA = X   + (size_t)(s + (half << 1)) * D  + dimc;  // rows K0|K2
  float*        qA = out + (size_t)(s + (half << 1)) * D2 + dim;

  int r0 = s;
  for (; r0 < efull; r0 += 4) {
    __builtin_prefetch(pA + stepX, 0, 1);            // global_prefetch_b8
    const int l0 = label[r0];
    const int l1 = label[r0 + 1];
    const int l2 = label[r0 + 2];
    const int l3 = label[r0 + 3];
    const float w0 = (l0 == -1) ? 1.f : 0.f;
    const float w1 = (l1 == -1) ? 1.f : 0.f;
    const float w2 = (l2 == -1) ? 1.f : 0.f;
    const float w3 = (l3 == -1) ? 1.f : 0.f;
    cntf += countf * (w0 + w1 + w2 + w3);

    const float b0v = __builtin_nontemporal_load(pA);
    const float b1v = __builtin_nontemporal_load(pA + D);

    v2f Av; Av.x = half ? w2 : w0;  Av.y = half ? w3 : w1;  // mask over K (rows), same for all M
    v2f Bv; Bv.x = b0v;             Bv.y = b1v;             // rows x dims

    acc = __builtin_amdgcn_wmma_f32_16x16x4_f32(
        false, Av, false, Bv, (short)0, acc, false, false);

    if (FULL) {                                   // common path: no EXEC dance
      __builtin_nontemporal_store(b0v, qA);
      __builtin_nontemporal_store(b1v, qA + D2);
    } else if (dok) {
      __builtin_nontemporal_store(b0v, qA);
      __builtin_nontemporal_store(b1v, qA + D2);
    }
    pA += stepX;
    qA += stepO;
  }

  if (r0 < e) {                                   // peeled tail: 1-3 rows
    const int l0 = label[r0];
    const int l1 = (r0 + 1 < e) ? label[r0 + 1] : 0;
    const int l2 = (r0 + 2 < e) ? label[r0 + 2] : 0;
    const float w0 = (l0 == -1) ? 1.f : 0.f;
    const float w1 = (l1 == -1) ? 1.f : 0.f;
    const float w2 = (l2 == -1) ? 1.f : 0.f;
    cntf += countf * (w0 + w1 + w2);

    const int rA  = r0 + (half << 1);
    const int rB  = rA + 1;
    const int rAc = (rA < e) ? rA : (e - 1);      // clamp: weight is 0 anyway
    const int rBc = (rB < e) ? rB : (e - 1);
    const float b0v = __builtin_nontemporal_load(X + (size_t)rAc * D + dimc);
    const float b1v = __builtin_nontemporal_load(X + (size_t)rBc * D + dimc);

    v2f Av; Av.x = half ? w2 : w0;  Av.y = half ? 0.f : w1;
    v2f Bv; Bv.x = b0v;             Bv.y = b1v;

    acc = __builtin_amdgcn_wmma_f32_16x16x4_f32(   // full EXEC here
        false, Av, false, Bv, (short)0, acc, false, false);

    if (dok && rA < e)
      __builtin_nontemporal_store(b0v, out + (size_t)rA * D2 + dim);
    if (dok && rB < e)
      __builtin_nontemporal_store(b1v, out + (size_t)rB * D2 + dim);
  }

  // C layout: VGPR0, lanes 0-15 = (M=0, N=lane); all M rows identical here.
  if (half == 0 && dok) pool[dim] = acc[0];
}

// ---------------------------------------------------------------------------
// Kernel 2: fused per-graph kernel, one workgroup (10 waves) per graph.
// Reads node_rep once, writes both halves of out once: 1.44 GB total.
// ---------------------------------------------------------------------------
__launch_bounds__(320)
__global__ void fused_cond_pool_kernel(const float* __restrict__ X,
                                       const int* __restrict__ label,
                                       const int* __restrict__ numg,
                                       const int* __restrict__ start,
                                       float* __restrict__ out,
                                       int D, int N) {
  const int B      = *numg;
  const int D2     = 2 * D;
  const int nch    = (D + 15) >> 4;
  const int tid    = threadIdx.x;
  const int wave   = tid >> 5;
  const int lane   = tid & 31;
  const int half   = lane >> 4;
  const int nwaves = blockDim.x >> 5;

  __shared__ f4    pool4[256];                    // raw sums, D <= 1024
  __shared__ float s_scale;
  float* pool = (float*)pool4;

  for (int g = blockIdx.x; g < B; g += gridDim.x) {
    const int s = start[g];
    const int e = start[g + 1];
    if (s == e) continue;                         // uniform: empty graph

    float cntf = 0.f;
    for (int ch = wave; ch < nch; ch += nwaves) {
      const int  dbase  = ch << 4;
      const int  dim    = dbase + (lane & 15);
      const bool full   = (dbase + 16 <= D);
      const bool dok    = dim < D;
      const int  dimc   = dok ? dim : (D - 1);
      const float countf = (ch == wave) ? 1.f : 0.f;  // count rows once per wave
      if (full)
        process_chunk<true >(X, label, out, pool, s, e, D, D2,
                             dim, dimc, dok, half, countf, cntf);
      else
        process_chunk<false>(X, label, out, pool, s, e, D, D2,
                             dim, dimc, dok, half, countf, cntf);
    }

    if (tid == 0) {                               // wave 0 holds the count
      const bool fl = (label[e - 1] == -1);       // last-node gate
      s_scale = fl ? (1.f / fmaxf(cntf, 1.f)) : 0.f;
    }
    __syncthreads();

    // broadcast gated mean into second halves: 4 rows in parallel, f4 NT stores
    const int   D4 = D >> 2;
    const int   c  = tid % 80;
    const int   rp = tid / 80;
    const float sc = s_scale;
    f4 pv = {};
    if (c < D4) { pv = pool4[c]; pv *= sc; }
    for (int n = s + rp; n < e; n += 4) {
      if (c < D4)
        __builtin_nontemporal_store(pv, (f4*)(out + (size_t)n * D2 + D) + c);
    }
    if (D & 3) {                                  // scalar remainder (unused for D=300)
      const int dr = (D4 << 2) + c;
      float rv = (dr < D) ? pool[dr] * sc : 0.f;
      for (int n = s + rp; n < e; n += 4)
        if (dr < D)
          __builtin_nontemporal_store(rv, out + (size_t)n * D2 + D + dr);
    }
    __syncthreads();                              // pool/scale reused next graph
  }
}

// ---------------------------------------------------------------------------
extern "C" void kernel_launch(void* const* d_in, const int* in_sizes, int n_in,
                              void* d_out, int out_size, void* d_ws, size_t ws_size,
                              hipStream_t stream) {
  const float* X     = (const float*)d_in[0];   // node_rep [N, D] f32
  const int*   batch = (const int*)d_in[1];     // [N] sorted
  const int*   label = (const int*)d_in[2];     // [N]
  const int*   numg  = (const int*)d_in[3];     // scalar B (device)
  float*       out   = (float*)d_out;           // [N, 2D] f32

  const int N = in_sizes[1];
  const int D = in_sizes[0] / N;
  int* start = (int*)d_ws;                      // (B+1) ints of scratch

  graph_bounds_kernel<<<64, 256, 0, stream>>>(batch, numg, N, start);
  fused_cond_pool_kernel<<<4096, 320, 0, stream>>>(X, label, numg, start, out, D, N);
}